// GraphSAGE_14594298872526
// MI455X (gfx1250) — compile-verified
//
#include <hip/hip_runtime.h>
#include <stdint.h>

// GraphSAGE (max aggregation), 2 layers, on MI455X (gfx1250, wave32).
// Scatter-max via native u32 atomics in L2 (whole working set < 192MB L2),
// layer-1 GEMM via V_WMMA_F32_16X16X4_F32 (exact fp32), layer-2 via VALU.

#define N_NODES 100000
#define IN_C    50
#define HID_C   64
#define OUT_C   2
#define KPAD    52   // K padded to multiple of 4 for the 16x16x4 WMMA

typedef float v2f __attribute__((ext_vector_type(2)));
typedef float v8f __attribute__((ext_vector_type(8)));

// Monotone order-preserving float <-> uint mapping (branchless):
//   f1 < f2  <=>  map(f1) < map(f2)  (unsigned compare)
// map(f) >= 0x007FFFFF for any non-NaN float, so 0 is a safe "empty segment"
// sentinel that unmaps to 0.0f (exactly PyG's fill for isolated nodes).
__device__ __forceinline__ unsigned mapf(float f) {
  unsigned u = __float_as_uint(f);
  // sign set -> ~u ; sign clear -> u | 0x80000000   (3 VALU ops)
  return u ^ (((unsigned)((int)u >> 31)) | 0x80000000u);
}
__device__ __forceinline__ float unmapf(unsigned m) {
  // sign set -> m ^ 0x80000000 ; sign clear -> ~m   (branchless)
  unsigned u = m ^ ((~((unsigned)((int)m >> 31))) | 0x80000000u);
  u = (m == 0u) ? 0u : u;            // empty-segment sentinel -> +0.0f
  return __uint_as_float(u);
}

// ---------------------------------------------------------------- init agg
__global__ void k_init(unsigned* __restrict__ agg, int n4) {
  int i = blockIdx.x * blockDim.x + threadIdx.x;
  if (i < n4) ((uint4*)agg)[i] = make_uint4(0u, 0u, 0u, 0u);
}

// ------------------------------------------------------------ edge scatter-max
// 64 consecutive lanes handle the channels of one edge: coalesced feature
// reads and coalesced GLOBAL_ATOMIC_MAX_U32 into agg[dst][c]. The edge id is
// wave-uniform (64 lanes/edge, waves are 32-aligned), so src/dst go through
// readfirstlane -> scalar (SMEM) index loads. All indexing is 32-bit.
__global__ void k_scatter_max(const float* __restrict__ feat,
                              const int* __restrict__ ei,   // [2, E]
                              unsigned* __restrict__ agg,   // [N, HID_C] mapped
                              int E, int C) {
  int t = blockIdx.x * blockDim.x + threadIdx.x;            // < E*64 <= 102.4M
  int e = __builtin_amdgcn_readfirstlane(t >> 6);           // wave-uniform edge id
  int c = t & 63;
  if (e >= E || c >= C) return;
  int s = ei[e];                                            // scalar loads
  int d = ei[E + e];
  float v = feat[s * C + c];
  atomicMax(&agg[d * HID_C + c], mapf(v));
}

// ------------------------------------------------------- layer-1 WMMA GEMM
// h = relu( unmap(agg) @ W1l^T + b1 + x @ W1r^T ),   [N,50/64] -> [N,64]
// One wave per 16-node tile; 4 f32 accumulators cover the 64 output columns;
// both paths (agg*Wl and x*Wr) accumulate into the same C. Weights live in
// LDS pre-transposed to B layout with stride KPAD=52 (zero-padded k>=50):
// conflict-free ds_load_b64 fragments.
__global__ void __launch_bounds__(256)
k_gemm1_wmma(const unsigned* __restrict__ agg, const float* __restrict__ x,
             const float* __restrict__ W1l, const float* __restrict__ b1,
             const float* __restrict__ W1r, float* __restrict__ h) {
  __shared__ float sWl[HID_C * KPAD];
  __shared__ float sWr[HID_C * KPAD];
  const int tid = threadIdx.x;
  for (int idx = tid; idx < HID_C * KPAD; idx += 256) {
    int n = idx / KPAD;
    int k = idx - n * KPAD;
    sWl[idx] = (k < IN_C) ? W1l[n * IN_C + k] : 0.0f;
    sWr[idx] = (k < IN_C) ? W1r[n * IN_C + k] : 0.0f;
  }
  __syncthreads();

  const int wave = tid >> 5;
  const int lane = tid & 31;
  const int half = lane >> 4;   // A frag: K pair (k0,k0+1) vs (k0+2,k0+3); C/D: rows 0-7 vs 8-15
  const int l16  = lane & 15;

  const int tile = blockIdx.x * 8 + wave;     // 16-row tile id
  const int row0 = tile * 16;
  if (row0 >= N_NODES) return;                // N % 16 == 0 -> whole-wave uniform exit

  const int m = row0 + l16;                   // A-matrix row held by this lane
  const unsigned* aggRow = agg + m * HID_C;
  const float*    xRow   = x   + m * IN_C;

  v8f acc[4];
#pragma unroll
  for (int i = 0; i < 4; ++i)
#pragma unroll
    for (int j = 0; j < 8; ++j) acc[i][j] = 0.0f;

#pragma unroll
  for (int k0 = 0; k0 < KPAD; k0 += 4) {
    const int ka = k0 + 2 * half;             // this lane's K pair

    // A fragment, aggregation path (agg is padded: cols >= 50 are mapped-0 -> 0.0f)
    uint2 ua = *(const uint2*)(aggRow + ka);
    v2f a_agg;
    a_agg.x = unmapf(ua.x);
    a_agg.y = unmapf(ua.y);

    // A fragment, root path (x rows are only 50 wide -> zero-pad k in [50,52))
    v2f a_x;
    if (ka < IN_C) {
      float2 fx = *(const float2*)(xRow + ka);
      a_x.x = fx.x; a_x.y = fx.y;
    } else {
      a_x.x = 0.0f; a_x.y = 0.0f;
    }

#pragma unroll
    for (int nt = 0; nt < 4; ++nt) {
      const int n = nt * 16 + l16;
      v2f bl = *(const v2f*)(&sWl[n * KPAD + ka]);
      v2f br = *(const v2f*)(&sWr[n * KPAD + ka]);
      // D = A*B + C, exact fp32; both paths share one accumulator.
      acc[nt] = __builtin_amdgcn_wmma_f32_16x16x4_f32(
          false, a_agg, false, bl, (short)0, acc[nt], false, false);
      acc[nt] = __builtin_amdgcn_wmma_f32_16x16x4_f32(
          false, a_x,   false, br, (short)0, acc[nt], false, false);
    }
  }

  // Epilogue: bias + ReLU, store h[row][col] (16 consecutive cols per half-wave).
#pragma unroll
  for (int nt = 0; nt < 4; ++nt) {
    const int col = nt * 16 + l16;
    const float bias = b1[col];
#pragma unroll
    for (int v = 0; v < 8; ++v) {
      const int r = row0 + v + 8 * half;      // C/D layout: M = v + 8*half
      float val = acc[nt][v] + bias;
      h[r * HID_C + col] = val > 0.0f ? val : 0.0f;
    }
  }
}

// --------------------------------------------------------------- layer 2
// out[n][0..1] = unmap(agg2[n]) @ W2l^T + b2 + h[n] @ W2r^T   (OUT_C = 2)
__global__ void k_layer2(const unsigned* __restrict__ agg, const float* __restrict__ h,
                         const float* __restrict__ W2l, const float* __restrict__ b2,
                         const float* __restrict__ W2r, float* __restrict__ out) {
  int n = blockIdx.x * blockDim.x + threadIdx.x;
  if (n >= N_NODES) return;
  const unsigned* ar = agg + n * HID_C;
  const float*    hr = h   + n * HID_C;
  float a0 = 0.0f, a1 = 0.0f;
#pragma unroll 8
  for (int k = 0; k < HID_C; ++k) {
    float a  = unmapf(ar[k]);
    float hh = hr[k];
    a0 += a * W2l[k]         + hh * W2r[k];
    a1 += a * W2l[HID_C + k] + hh * W2r[HID_C + k];
  }
  out[n * OUT_C + 0] = a0 + b2[0];
  out[n * OUT_C + 1] = a1 + b2[1];
}

// ---------------------------------------------------------------- launch
extern "C" void kernel_launch(void* const* d_in, const int* in_sizes, int n_in,
                              void* d_out, int out_size, void* d_ws, size_t ws_size,
                              hipStream_t stream) {
  const float* x   = (const float*)d_in[0];
  const int*   ei  = (const int*)  d_in[1];
  const float* W1l = (const float*)d_in[2];
  const float* b1  = (const float*)d_in[3];
  const float* W1r = (const float*)d_in[4];
  const float* W2l = (const float*)d_in[5];
  const float* b2  = (const float*)d_in[6];
  const float* W2r = (const float*)d_in[7];
  float* out = (float*)d_out;

  const int E = in_sizes[1] / 2;

  // Workspace: agg (mapped u32, [N,64]) then h (f32, [N,64]) -> 51.2 MB total.
  unsigned* agg = (unsigned*)d_ws;
  float*    h   = (float*)((char*)d_ws + (size_t)N_NODES * HID_C * sizeof(unsigned));

  const dim3 b256(256);
  const int  initN4     = N_NODES * HID_C / 4;
  const int  initBlocks = (initN4 + 255) / 256;
  const long long sth   = (long long)E * 64;
  const int  sBlocks    = (int)((sth + 255) / 256);
  const int  gBlocks    = (N_NODES / 16 + 7) / 8;           // 6250 tiles, 8 waves/block
  const int  oBlocks    = (N_NODES + 255) / 256;

  // Layer 1
  k_init<<<initBlocks, b256, 0, stream>>>(agg, initN4);
  k_scatter_max<<<sBlocks, b256, 0, stream>>>(x, ei, agg, E, IN_C);
  k_gemm1_wmma<<<gBlocks, b256, 0, stream>>>(agg, x, W1l, b1, W1r, h);

  // Layer 2 (reuse agg buffer)
  k_init<<<initBlocks, b256, 0, stream>>>(agg, initN4);
  k_scatter_max<<<sBlocks, b256, 0, stream>>>(h, ei, agg, E, HID_C);
  k_layer2<<<oBlocks, b256, 0, stream>>>(agg, h, W2l, b2, W2r, out);
}